// PillarEncoder_84868553769572
// MI455X (gfx1250) — compile-verified
//
#include <hip/hip_runtime.h>
#include <math.h>

typedef __attribute__((ext_vector_type(2))) float v2f;
typedef __attribute__((ext_vector_type(8))) float v8f;

#define P_MAX 32
#define IN_CH 9
#define KPAD 12
#define OUT_CH 64
#define XL 432
#define YL 496
#define H_IMG 375
#define W_IMG 1242
#define VXc 0.16f
#define VYc 0.16f
#define X_OFFc 0.08f
#define Y_OFFc (-39.6f)
#define BN_EPSc 1e-3f
#define WAVES_PER_BLOCK 4

__global__ __launch_bounds__(128) void pillar_encoder_kernel(
    const float* __restrict__ pillars,      // (N, 32, 4)
    const int*   __restrict__ coors,        // (N, 3)
    const int*   __restrict__ npoints,      // (N,)
    const float* __restrict__ prob_maps,    // (B, 4, 375, 1242)
    const float* __restrict__ calibs,       // (B, 3, 4)
    const float* __restrict__ conv_w,       // (64, 9)
    const float* __restrict__ bn_gamma,
    const float* __restrict__ bn_beta,
    const float* __restrict__ bn_mean,
    const float* __restrict__ bn_var,
    const float* __restrict__ sp_w1,        // (32, 4)
    const float* __restrict__ sp_b1,        // (32,)
    const float* __restrict__ sp_w2,        // (64, 32)
    const float* __restrict__ sp_b2,        // (64,)
    const float* __restrict__ gf_w,         // (64, 68)
    const float* __restrict__ gf_b,         // (64,)
    float* __restrict__ out,                // (B, 64, 496, 432), pre-zeroed
    int N)
{
    __shared__ float ldsFeat[WAVES_PER_BLOCK][P_MAX * KPAD];
    __shared__ float ldsPooled[WAVES_PER_BLOCK][OUT_CH];
    __shared__ float ldsHid[WAVES_PER_BLOCK][32];

    const int lane = threadIdx.x & 31;
    const int wave = threadIdx.x >> 5;
    int pillar = blockIdx.x * WAVES_PER_BLOCK + wave;
    if (pillar >= N) pillar = N - 1;   // uniform control flow; duplicate work is benign

    // ---- load this lane's point (lane == point index) ----
    const float4 pt = ((const float4*)pillars)[(size_t)pillar * P_MAX + lane];
    const int np = npoints[pillar];
    const int b  = coors[pillar * 3 + 0];
    const int ix = coors[pillar * 3 + 1];
    const int iy = coors[pillar * 3 + 2];

    // ---- mean center: sum over ALL 32 points, divide by npoints (matches ref) ----
    float sx = pt.x, sy = pt.y, sz = pt.z;
    #pragma unroll
    for (int off = 16; off > 0; off >>= 1) {
        sx += __shfl_xor(sx, off, 32);
        sy += __shfl_xor(sy, off, 32);
        sz += __shfl_xor(sz, off, 32);
    }
    const float inv_np = 1.0f / (float)np;
    const float mcx = sx * inv_np, mcy = sy * inv_np, mcz = sz * inv_np;

    // ---- 9-channel features, masked, staged to LDS (K padded to 12) ----
    const float xo = pt.x - ((float)ix * VXc + X_OFFc);
    const float yo = pt.y - ((float)iy * VYc + Y_OFFc);
    const float m  = (lane < np) ? 1.0f : 0.0f;
    {
        float* f = &ldsFeat[wave][lane * KPAD];
        f[0] = xo * m;
        f[1] = yo * m;
        f[2] = pt.z * m;
        f[3] = pt.w * m;
        f[4] = (pt.x - mcx) * m;
        f[5] = (pt.y - mcy) * m;
        f[6] = (pt.z - mcz) * m;
        f[7] = xo * m;
        f[8] = yo * m;
        f[9] = 0.0f; f[10] = 0.0f; f[11] = 0.0f;
    }

    // ---- projection + prob gather (redundant on all lanes: avoids intra-wave sync) ----
    const float* C = calibs + b * 12;
    const float w0 = C[0]*mcx + C[1]*mcy + C[2]*mcz  + C[3];
    const float w1 = C[4]*mcx + C[5]*mcy + C[6]*mcz  + C[7];
    const float w2 = C[8]*mcx + C[9]*mcy + C[10]*mcz + C[11];
    float uf = w0 / w2, vf = w1 / w2;
    uf = fminf(fmaxf(uf, 0.0f), (float)(W_IMG - 1));
    vf = fminf(fmaxf(vf, 0.0f), (float)(H_IMG - 1));
    const int u = (int)uf, v = (int)vf;
    float prob[4];
    #pragma unroll
    for (int j = 0; j < 4; ++j)
        prob[j] = prob_maps[(((size_t)b * 4 + j) * H_IMG + v) * W_IMG + u];

    // ---- sem MLP hidden layer: one hidden unit per lane -> LDS ----
    {
        float h = sp_b1[lane];
        #pragma unroll
        for (int j = 0; j < 4; ++j) h += prob[j] * sp_w1[lane * 4 + j];
        ldsHid[wave][lane] = fmaxf(h, 0.0f);
    }

    // ---- B operands: conv_w^T as 4x16 f32 tiles (lanes 0-15: K0/K1, 16-31: K2/K3) ----
    const int col = lane & 15;
    const int hi  = (lane >> 4) & 1;
    v2f Bop[4][3];
    float bscale[4], bshift[4];
    #pragma unroll
    for (int n = 0; n < 4; ++n) {
        const int o = n * 16 + col;
        #pragma unroll
        for (int k = 0; k < 3; ++k) {
            const int c0 = 4 * k + 2 * hi;
            Bop[n][k].x = (c0     < IN_CH) ? conv_w[o * IN_CH + c0]     : 0.0f;
            Bop[n][k].y = (c0 + 1 < IN_CH) ? conv_w[o * IN_CH + c0 + 1] : 0.0f;
        }
        const float s = bn_gamma[o] * rsqrtf(bn_var[o] + BN_EPSc);
        bscale[n] = s;
        bshift[n] = bn_beta[o] - bn_mean[o] * s;
    }

    __syncthreads();   // feats visible; uniform across the block

    // ---- GEMM via V_WMMA_F32_16X16X4_F32: 2 M-tiles x 4 N-tiles x 3 K-steps ----
    float colmax[4] = {0.0f, 0.0f, 0.0f, 0.0f};
    #pragma unroll
    for (int t = 0; t < 2; ++t) {
        v2f Aop[3];
        const int mrow = t * 16 + col;
        #pragma unroll
        for (int k = 0; k < 3; ++k) {
            const int c0 = 4 * k + 2 * hi;
            Aop[k].x = ldsFeat[wave][mrow * KPAD + c0];
            Aop[k].y = ldsFeat[wave][mrow * KPAD + c0 + 1];
        }
        #pragma unroll
        for (int n = 0; n < 4; ++n) {
            v8f acc = {0.f, 0.f, 0.f, 0.f, 0.f, 0.f, 0.f, 0.f};
            #pragma unroll
            for (int k = 0; k < 3; ++k) {
                acc = __builtin_amdgcn_wmma_f32_16x16x4_f32(
                        false, Aop[k], false, Bop[n][k],
                        (short)0, acc, false, false);
            }
            // BN + ReLU + max over the 8 M-rows this lane holds
            float mx = colmax[n];
            #pragma unroll
            for (int i = 0; i < 8; ++i) {
                float h = acc[i] * bscale[n] + bshift[n];
                mx = fmaxf(mx, fmaxf(h, 0.0f));
            }
            colmax[n] = mx;
        }
    }

    // combine half-waves (rows 0-7 vs 8-15 of each tile) -> pooled per channel
    #pragma unroll
    for (int n = 0; n < 4; ++n) {
        const float p = fmaxf(colmax[n], __shfl_xor(colmax[n], 16, 32));
        if (lane < 16) ldsPooled[wave][n * 16 + lane] = p;
    }

    __syncthreads();   // pooled + hid visible

    // ---- gate + sem + fuse + scatter (2 output channels per lane) ----
    #pragma unroll
    for (int half = 0; half < 2; ++half) {
        const int o = lane + 32 * half;
        const float* gw = gf_w + o * 68;
        float g = gf_b[o];
        #pragma unroll 8
        for (int c = 0; c < OUT_CH; ++c) g += ldsPooled[wave][c] * gw[c];
        #pragma unroll
        for (int j = 0; j < 4; ++j) g += prob[j] * gw[64 + j];
        g = 1.0f / (1.0f + __expf(-g));

        const float* w2p = sp_w2 + o * 32;
        float sm = sp_b2[o];
        #pragma unroll 8
        for (int h = 0; h < 32; ++h) sm += ldsHid[wave][h] * w2p[h];

        const float pl = ldsPooled[wave][o];
        const float fused = pl * g + sm * (1.0f - g);
        out[(((size_t)b * OUT_CH + o) * YL + iy) * XL + ix] = fused;
    }
}

extern "C" void kernel_launch(void* const* d_in, const int* in_sizes, int n_in,
                              void* d_out, int out_size, void* d_ws, size_t ws_size,
                              hipStream_t stream) {
    const float* pillars   = (const float*)d_in[0];
    const int*   coors     = (const int*)  d_in[1];
    const int*   npts      = (const int*)  d_in[2];
    const float* prob_maps = (const float*)d_in[3];
    const float* calibs    = (const float*)d_in[4];
    const float* conv_w    = (const float*)d_in[5];
    const float* bn_gamma  = (const float*)d_in[6];
    const float* bn_beta   = (const float*)d_in[7];
    const float* bn_mean   = (const float*)d_in[8];
    const float* bn_var    = (const float*)d_in[9];
    const float* sp_w1     = (const float*)d_in[10];
    const float* sp_b1     = (const float*)d_in[11];
    const float* sp_w2     = (const float*)d_in[12];
    const float* sp_b2     = (const float*)d_in[13];
    const float* gf_w      = (const float*)d_in[14];
    const float* gf_b      = (const float*)d_in[15];
    float* out = (float*)d_out;

    const int N = in_sizes[2];   // number of pillars (npoints_per_pillar count)

    // Canvas zero-fill: the dominant memory traffic (~219 MB), capture-legal memset.
    hipMemsetAsync(d_out, 0, (size_t)out_size * sizeof(float), stream);

    const int blocks = (N + WAVES_PER_BLOCK - 1) / WAVES_PER_BLOCK;
    pillar_encoder_kernel<<<blocks, WAVES_PER_BLOCK * 32, 0, stream>>>(
        pillars, coors, npts, prob_maps, calibs, conv_w,
        bn_gamma, bn_beta, bn_mean, bn_var,
        sp_w1, sp_b1, sp_w2, sp_b2, gf_w, gf_b, out, N);
}